// GraphSAGE_41558103556324
// MI455X (gfx1250) — compile-verified
//
#include <hip/hip_runtime.h>

typedef __attribute__((ext_vector_type(2))) float v2f;
typedef __attribute__((ext_vector_type(8))) float v8f;

#define NNODES 50000
#define D 128
#define EPSBN 1e-5f

// ---------------- utility: zero a float buffer ----------------
__global__ void k_zero(float* __restrict__ p, int n) {
  int i = blockIdx.x * blockDim.x + threadIdx.x;
  int stride = gridDim.x * blockDim.x;
  for (; i < n; i += stride) p[i] = 0.0f;
}

// ---------------- degree count: cnt[dst] += 1 ----------------
__global__ void k_count(const long long* __restrict__ dst, float* __restrict__ cnt, int E) {
  int e = blockIdx.x * blockDim.x + threadIdx.x;
  if (e < E) atomicAdd(&cnt[(int)dst[e]], 1.0f);
}

// ------- scatter-add: agg[dst] += x[src], 32 lanes per edge, 4 floats/lane -------
__global__ void k_scatter(const long long* __restrict__ src, const long long* __restrict__ dst,
                          const float* __restrict__ X, float* __restrict__ agg, int E) {
  long long tid = (long long)blockIdx.x * blockDim.x + threadIdx.x;
  int e    = (int)(tid >> 5);
  int lane = (int)(tid & 31);
  if (e >= E) return;
  int s = (int)src[e];
  int d = (int)dst[e];
  const float4 v = *(const float4*)(X + (long long)s * D + lane * 4);
  float* o = agg + (long long)d * D + lane * 4;
  atomicAdd(o + 0, v.x);
  atomicAdd(o + 1, v.y);
  atomicAdd(o + 2, v.z);
  atomicAdd(o + 3, v.w);
}

// ------- fused SAGE GEMM: H = (agg / max(cnt,1)) @ Wl + X @ Wr + bias -------
// Block = 8 waves sharing one N-tile; the 8 KB x2 weight tiles are staged in LDS
// once per block and re-read with ds_load. One wave per 16x16 output tile,
// K=128 -> 64 V_WMMA_F32_16X16X4_F32 per tile (both matmuls interleaved).
__global__ void k_sage_gemm(const float* __restrict__ Agg, const float* __restrict__ cnt,
                            const float* __restrict__ X,
                            const float* __restrict__ Wl, const float* __restrict__ Wr,
                            const float* __restrict__ bias,
                            float* __restrict__ H, int M) {
  __shared__ float Bs[2][D * 16];            // [matrix][k*16 + col], 16 KB

  const int tn = blockIdx.y;                 // N tile (0..7), shared by whole block

  // ---- stage Wl/Wr column tiles into LDS: thread t handles (matrix t>>7, row t&127)
  {
    const int t    = threadIdx.x;
    const int mat  = t >> 7;                 // 0 = Wl, 1 = Wr
    const int krow = t & (D - 1);
    const float4* wsrc = (const float4*)((mat ? Wr : Wl) + (long long)krow * D + tn * 16);
    float4* wdst = (float4*)&Bs[mat][krow * 16];
    wdst[0] = wsrc[0];
    wdst[1] = wsrc[1];
    wdst[2] = wsrc[2];
    wdst[3] = wsrc[3];
  }
  __syncthreads();

  const int wib  = threadIdx.x >> 5;         // wave in block (0..7)
  const int tm   = blockIdx.x * 8 + wib;     // M tile
  if (tm < M / 16) {                         // 50000 = 3125*16, last block ragged
    const int lane = threadIdx.x & 31;
    const int half = lane >> 4;              // 0: K pair {0,1}, 1: K pair {2,3}
    const int l15  = lane & 15;

    const int arow = tm * 16 + l15;          // A row for this lane (both halves)
    const int bcol = tn * 16 + l15;          // B / D column for this lane
    const float sc = 1.0f / fmaxf(cnt[arow], 1.0f);   // mean-aggregation scale

    v8f c = {0.f, 0.f, 0.f, 0.f, 0.f, 0.f, 0.f, 0.f};

    const float* Ap = Agg + (long long)arow * D + 2 * half;
    const float* Xp = X   + (long long)arow * D + 2 * half;

    #pragma unroll 8
    for (int k0 = 0; k0 < D; k0 += 4) {
      const int kb = k0 + 2 * half;
      // matmul 1: mean-aggregated neighbors @ Wl
      {
        float2 av = *(const float2*)(Ap + k0);
        v2f a, b;
        a.x = av.x * sc;
        a.y = av.y * sc;
        b.x = Bs[0][kb * 16 + l15];
        b.y = Bs[0][(kb + 1) * 16 + l15];
        c = __builtin_amdgcn_wmma_f32_16x16x4_f32(false, a, false, b, (short)0, c, false, false);
      }
      // matmul 2: self features @ Wr
      {
        float2 xv = *(const float2*)(Xp + k0);
        v2f a, b;
        a.x = xv.x;
        a.y = xv.y;
        b.x = Bs[1][kb * 16 + l15];
        b.y = Bs[1][(kb + 1) * 16 + l15];
        c = __builtin_amdgcn_wmma_f32_16x16x4_f32(false, a, false, b, (short)0, c, false, false);
      }
    }

    const float bb = bias[bcol];
    #pragma unroll
    for (int r = 0; r < 8; ++r) {
      const int row = tm * 16 + r + 8 * half;   // C/D layout: lanes 16-31 hold M = r+8
      H[(long long)row * D + bcol] = c[r] + bb;
    }
  }
}

// ------- BN column statistics: stats[c] = sum, stats[D+c] = sum of squares -------
__global__ void k_bn_stats(const float* __restrict__ H, float* __restrict__ stats, int M) {
  const int c = threadIdx.x;  // blockDim.x == D
  float s = 0.0f, q = 0.0f;
  for (int r = blockIdx.x; r < M; r += gridDim.x) {
    const float v = H[(long long)r * D + c];
    s += v;
    q += v * v;
  }
  atomicAdd(&stats[c], s);
  atomicAdd(&stats[D + c], q);
}

// ------- BN coefficients: scale = gamma*rsqrt(var+eps), shift = beta - mu*scale -------
__global__ void k_bn_coef(const float* __restrict__ stats, const float* __restrict__ gamma,
                          const float* __restrict__ beta, float* __restrict__ scale,
                          float* __restrict__ shift, int M) {
  const int c = threadIdx.x;  // blockDim.x == D
  const float mu  = stats[c] / (float)M;
  const float var = stats[D + c] / (float)M - mu * mu;  // biased variance, matches reference
  const float rs  = rsqrtf(var + EPSBN) * gamma[c];
  scale[c] = rs;
  shift[c] = beta[c] - mu * rs;
}

// ------- fused BN apply + ReLU, in place -------
__global__ void k_bn_relu(float* __restrict__ H, const float* __restrict__ scale,
                          const float* __restrict__ shift, long long n) {
  long long i = (long long)blockIdx.x * blockDim.x + threadIdx.x;
  const long long stride = (long long)gridDim.x * blockDim.x;
  for (; i < n; i += stride) {
    const int c = (int)(i & (D - 1));
    H[i] = fmaxf(H[i] * scale[c] + shift[c], 0.0f);
  }
}

// ------- output projection: out = H @ W_out(128x2) + b_out -------
__global__ void k_out(const float* __restrict__ H, const float* __restrict__ Wout,
                      const float* __restrict__ bout, float* __restrict__ out, int M) {
  const int r = blockIdx.x * blockDim.x + threadIdx.x;
  if (r >= M) return;
  float a0 = 0.0f, a1 = 0.0f;
  const float4* hp = (const float4*)(H + (long long)r * D);
  #pragma unroll 8
  for (int k4 = 0; k4 < D / 4; ++k4) {
    const float4 h = hp[k4];
    const int k = k4 * 4;
    a0 += h.x * Wout[(k + 0) * 2] + h.y * Wout[(k + 1) * 2] +
          h.z * Wout[(k + 2) * 2] + h.w * Wout[(k + 3) * 2];
    a1 += h.x * Wout[(k + 0) * 2 + 1] + h.y * Wout[(k + 1) * 2 + 1] +
          h.z * Wout[(k + 2) * 2 + 1] + h.w * Wout[(k + 3) * 2 + 1];
  }
  out[r * 2 + 0] = a0 + bout[0];
  out[r * 2 + 1] = a1 + bout[1];
}

extern "C" void kernel_launch(void* const* d_in, const int* in_sizes, int n_in,
                              void* d_out, int out_size, void* d_ws, size_t ws_size,
                              hipStream_t stream) {
  const float*     x    = (const float*)d_in[0];
  const long long* edge = (const long long*)d_in[1];   // int64 in reference
  const float* Wl0  = (const float*)d_in[2];
  const float* Wr0  = (const float*)d_in[3];
  const float* b0   = (const float*)d_in[4];
  const float* g0   = (const float*)d_in[5];
  const float* be0  = (const float*)d_in[6];
  const float* Wl1  = (const float*)d_in[7];
  const float* Wr1  = (const float*)d_in[8];
  const float* b1   = (const float*)d_in[9];
  const float* g1   = (const float*)d_in[10];
  const float* be1  = (const float*)d_in[11];
  const float* Wout = (const float*)d_in[12];
  const float* bout = (const float*)d_in[13];
  float* out = (float*)d_out;

  const int E = in_sizes[1] / 2;
  const long long MD = (long long)NNODES * D;   // 6,400,000 floats

  float* ws    = (float*)d_ws;
  float* agg   = ws;                 // 25.6 MB, reused by both layers
  float* h0    = ws + MD;            // 25.6 MB
  float* h1    = ws + 2 * MD;        // 25.6 MB
  float* cnt   = ws + 3 * MD;        // 50000 floats
  float* stats = cnt + NNODES;       // 2*D floats
  float* scale = stats + 2 * D;      // D floats
  float* shift = scale + D;          // D floats

  const long long* src = edge;
  const long long* dst = edge + E;

  const int mtiles = NNODES / 16;                           // 3125
  const dim3 gemmGrid((mtiles + 7) / 8, D / 16);            // 391 x 8 blocks
  const unsigned scatBlocks = (unsigned)(((long long)E * 32 + 255) / 256);

  // degrees (shared by both layers)
  k_zero<<<256, 256, 0, stream>>>(cnt, NNODES);
  k_count<<<(E + 255) / 256, 256, 0, stream>>>(dst, cnt, E);

  // ---------------- layer 0 ----------------
  k_zero<<<4096, 256, 0, stream>>>(agg, (int)MD);
  k_zero<<<1, 256, 0, stream>>>(stats, 2 * D);
  k_scatter<<<scatBlocks, 256, 0, stream>>>(src, dst, x, agg, E);
  k_sage_gemm<<<gemmGrid, 256, 0, stream>>>(agg, cnt, x, Wl0, Wr0, b0, h0, NNODES);
  k_bn_stats<<<512, D, 0, stream>>>(h0, stats, NNODES);
  k_bn_coef<<<1, D, 0, stream>>>(stats, g0, be0, scale, shift, NNODES);
  k_bn_relu<<<4096, 256, 0, stream>>>(h0, scale, shift, MD);

  // ---------------- layer 1 ----------------
  k_zero<<<4096, 256, 0, stream>>>(agg, (int)MD);
  k_zero<<<1, 256, 0, stream>>>(stats, 2 * D);
  k_scatter<<<scatBlocks, 256, 0, stream>>>(src, dst, h0, agg, E);
  k_sage_gemm<<<gemmGrid, 256, 0, stream>>>(agg, cnt, h0, Wl1, Wr1, b1, h1, NNODES);
  k_bn_stats<<<512, D, 0, stream>>>(h1, stats, NNODES);
  k_bn_coef<<<1, D, 0, stream>>>(stats, g1, be1, scale, shift, NNODES);
  k_bn_relu<<<4096, 256, 0, stream>>>(h1, scale, shift, MD);

  // ---------------- output projection ----------------
  k_out<<<(NNODES + 255) / 256, 256, 0, stream>>>(h1, Wout, bout, out, NNODES);
}